// VectorQuantize_85401129714120
// MI455X (gfx1250) — compile-verified
//
#include <hip/hip_runtime.h>

typedef __attribute__((ext_vector_type(2))) float v2f;
typedef __attribute__((ext_vector_type(4))) float v4f;
typedef __attribute__((ext_vector_type(8))) float v8f;

#define N_ROWS (16 * 4096)   // 65536 rows of x
#define DIMS 64
#define NEMB 1024
#define NTILES (NEMB / 16)   // 64 code tiles
#define ROWS_PER_WAVE 32     // two 16-row M tiles per wave
#define WAVES_PER_BLOCK 4
#define ROWS_PER_BLOCK (ROWS_PER_WAVE * WAVES_PER_BLOCK)  // 128
#define COMMIT_SCALE (1.25f / ((float)N_ROWS * (float)DIMS))

// ---------------------------------------------------------------------------
// Pass 1: per-code squared norms ||e||^2 into workspace.
// ---------------------------------------------------------------------------
__global__ void vq_enorm_kernel(const float* __restrict__ embed,
                                float* __restrict__ enorm) {
    int i = blockIdx.x * blockDim.x + threadIdx.x;
    if (i < NEMB) {
        const float* e = embed + (size_t)i * DIMS;
        float s = 0.f;
        #pragma unroll
        for (int d = 0; d < DIMS; ++d) s += e[d] * e[d];
        enorm[i] = s;
    }
}

// Issue one 4KB codebook tile (16 codes x 64 f32) into LDS asynchronously.
// 256 x b128 transfers; 128 threads issue 2 each. ASYNCcnt += 2 per wave.
__device__ __forceinline__ void stage_tile(const float* __restrict__ embed,
                                           int nb, float* dstLds, int tid) {
    const float* g0 = embed + (size_t)nb * DIMS + tid * 4;
    const float* g1 = g0 + 128 * 4;
    unsigned l0 = (unsigned)(uintptr_t)(dstLds + tid * 4);
    unsigned l1 = l0 + 128 * 16;
    asm volatile("global_load_async_to_lds_b128 %0, %1, off"
                 :: "v"(l0), "v"(g0) : "memory");
    asm volatile("global_load_async_to_lds_b128 %0, %1, off"
                 :: "v"(l1), "v"(g1) : "memory");
}

// ---------------------------------------------------------------------------
// Pass 2: fused distance-GEMM (WMMA f32 16x16x4) + argmin + gather + loss.
// Workgroup = 4 waves x 32 rows = 128 rows. Codebook tiles staged once per
// WG into triple-buffered LDS via async-to-LDS; each B fragment feeds two
// independent WMMA accumulator chains (2 M-tiles per wave).
// ---------------------------------------------------------------------------
__global__ void __launch_bounds__(WAVES_PER_BLOCK * 32)
vq_main_kernel(const float* __restrict__ x,
               const float* __restrict__ embed,
               const float* __restrict__ enorm,
               float* __restrict__ q_out,
               float* __restrict__ idx_out,
               float* __restrict__ loss_out) {
    __shared__ float bstage[3][16 * DIMS];   // 3 x 4KB

    const int tid    = threadIdx.x;
    const int lane   = tid & 31;
    const int waveIB = tid >> 5;
    const int row0   = blockIdx.x * ROWS_PER_BLOCK + waveIB * ROWS_PER_WAVE;

    const int lo     = lane & 15;   // position within half-wave
    const int hiHalf = lane >> 4;   // 0 or 1

    // A fragments for two M tiles (ISA 32-bit A 16x4 layout: lanes 0-15
    // carry M=lo, K={4k,4k+1}; lanes 16-31 carry M=lo, K={4k+2,4k+3}).
    v2f a0[16], a1[16];
    {
        const float* ap0 = x + (size_t)(row0 + lo) * DIMS + hiHalf * 2;
        const float* ap1 = ap0 + 16 * DIMS;
        #pragma unroll
        for (int k = 0; k < 16; ++k) {
            a0[k] = *(const v2f*)(ap0 + 4 * k);
            a1[k] = *(const v2f*)(ap1 + 4 * k);
        }
    }

    float bestd[2][8];
    int   besti[2][8];
    #pragma unroll
    for (int s = 0; s < 2; ++s)
        #pragma unroll
        for (int r = 0; r < 8; ++r) { bestd[s][r] = 3.4e38f; besti[s][r] = 0; }

    // Prime the async pipeline: tiles 0 and 1.
    stage_tile(embed, 0 * 16, bstage[0], tid);
    stage_tile(embed, 1 * 16, bstage[1], tid);

    for (int t = 0; t < NTILES; ++t) {
        // Wait for tile t's async transfers (FIFO: <=2 outstanding means the
        // oldest pair, i.e. tile t, has landed). Tail tile waits for all.
        if (t < NTILES - 1) {
            asm volatile("s_wait_asynccnt 0x2" ::: "memory");
        } else {
            asm volatile("s_wait_asynccnt 0x0" ::: "memory");
        }
        __syncthreads();   // tile t visible to all waves; buffer (t+2)%3 free

        if (t + 2 < NTILES)
            stage_tile(embed, (t + 2) * 16, bstage[(t + 2) % 3], tid);

        const float* bs = bstage[t % 3];
        const int code = t * 16 + lo;

        // B fragment (4x16 = embed^T tile), symmetric layout to A, from LDS.
        v2f b[16];
        #pragma unroll
        for (int k = 0; k < 16; ++k)
            b[k] = *(const v2f*)(bs + lo * DIMS + hiHalf * 2 + 4 * k);

        v8f acc0 = {}, acc1 = {};
        #pragma unroll
        for (int k = 0; k < 16; ++k) {
            acc0 = __builtin_amdgcn_wmma_f32_16x16x4_f32(
                false, a0[k], false, b[k], (short)0, acc0, false, false);
            acc1 = __builtin_amdgcn_wmma_f32_16x16x4_f32(
                false, a1[k], false, b[k], (short)0, acc1, false, false);
        }

        // dist' = ||e||^2 - 2 x.e  (||x||^2 constant per row, irrelevant)
        const float en = enorm[code];
        #pragma unroll
        for (int r = 0; r < 8; ++r) {
            float d0 = en - 2.0f * acc0[r];
            bool lt0 = (d0 < bestd[0][r]);
            bestd[0][r] = lt0 ? d0 : bestd[0][r];
            besti[0][r] = lt0 ? code : besti[0][r];
            float d1 = en - 2.0f * acc1[r];
            bool lt1 = (d1 < bestd[1][r]);
            bestd[1][r] = lt1 ? d1 : bestd[1][r];
            besti[1][r] = lt1 ? code : besti[1][r];
        }
    }

    // Lexicographic (dist, idx) min across the 16 lanes of each half-wave.
    // C/D layout: lanes 0-15 hold rows M=r (VGPR r), lanes 16-31 hold M=r+8.
    #pragma unroll
    for (int m = 1; m < 16; m <<= 1) {
        #pragma unroll
        for (int s = 0; s < 2; ++s)
            #pragma unroll
            for (int r = 0; r < 8; ++r) {
                float od = __shfl_xor(bestd[s][r], m, 32);
                int   oi = __shfl_xor(besti[s][r], m, 32);
                bool take = (od < bestd[s][r]) ||
                            (od == bestd[s][r] && oi < besti[s][r]);
                bestd[s][r] = take ? od : bestd[s][r];
                besti[s][r] = take ? oi : besti[s][r];
            }
    }

    // Output: each half-wave handles its 8 rows per M tile; 16 lanes cover
    // the 64 columns with float4s. Forward value of q_st == quantized code.
    float sumsq = 0.f;
    #pragma unroll
    for (int s = 0; s < 2; ++s) {
        #pragma unroll
        for (int r = 0; r < 8; ++r) {
            const int row = row0 + s * 16 + hiHalf * 8 + r;
            const int idx = besti[s][r];
            const float* ev = embed + (size_t)idx * DIMS + lo * 4;
            const float* xv = x + (size_t)row * DIMS + lo * 4;
            v4f e  = *(const v4f*)ev;
            v4f xx = *(const v4f*)xv;
            *(v4f*)(q_out + (size_t)row * DIMS + lo * 4) = e;
            v4f dd = xx - e;
            sumsq += dd.x * dd.x + dd.y * dd.y + dd.z * dd.z + dd.w * dd.w;
            if (lo == 0) idx_out[row] = (float)idx;
        }
    }

    // Wave-wide loss partial, one atomic per wave (2048 total).
    #pragma unroll
    for (int m = 1; m < 32; m <<= 1)
        sumsq += __shfl_xor(sumsq, m, 32);
    if (lane == 0)
        atomicAdd(loss_out, sumsq * COMMIT_SCALE);
}

// ---------------------------------------------------------------------------
extern "C" void kernel_launch(void* const* d_in, const int* in_sizes, int n_in,
                              void* d_out, int out_size, void* d_ws, size_t ws_size,
                              hipStream_t stream) {
    const float* x     = (const float*)d_in[0];   // [16, 4096, 64] fp32
    const float* embed = (const float*)d_in[1];   // [1024, 64] fp32

    float* q_out    = (float*)d_out;                          // 4194304
    float* idx_out  = q_out + (size_t)N_ROWS * DIMS;          // 65536
    float* loss_out = idx_out + N_ROWS;                       // 1
    float* enorm    = (float*)d_ws;                           // 1024 floats

    hipMemsetAsync(loss_out, 0, sizeof(float), stream);

    vq_enorm_kernel<<<NEMB / 256, 256, 0, stream>>>(embed, enorm);

    vq_main_kernel<<<N_ROWS / ROWS_PER_BLOCK, WAVES_PER_BLOCK * 32, 0, stream>>>(
        x, embed, enorm, q_out, idx_out, loss_out);
}